// SMSE_30056181137892
// MI455X (gfx1250) — compile-verified
//
#include <hip/hip_runtime.h>

// Problem: B=16, C=4, H=512, W=512  ->  64 planes of 262144 f32 px.
// Memory-bound masked-MSE reduction; WMMA f32 16x16x4 used for the
// wave-level cross-lane reduction (exact, uses the CDNA5 matrix pipe).

#define PLANES      64          // B*C
#define PLANE_ELEMS (512 * 512) // H*W = 262144
#define SUBBLOCKS   16          // blocks per plane
#define THREADS     256         // 8 waves (wave32)
#define PLANE_F4    (PLANE_ELEMS / 4)          // 65536 float4 per plane
#define CHUNK_F4    (PLANE_F4 / SUBBLOCKS)     // 4096 float4 per block

typedef float v2f __attribute__((ext_vector_type(2)));
typedef float v8f __attribute__((ext_vector_type(8)));

// Sum a per-lane f32 value across all 32 lanes of the wave using
// V_WMMA_F32_16X16X4_F32 with an all-ones B matrix.
// A layout (16x4 f32): lanes 0-15 VGPR0=K0, VGPR1=K1; lanes 16-31 VGPR0=K2, VGPR1=K3.
// With a = {v, 0}:  D[m][n] = v_m + v_{m+16}  (independent of n and of B layout,
// since B is all ones). D layout: lane n<16 VGPRj = D[j][n]; lane n>=16 VGPRj = D[j+8][n-16].
// Summing the 8 D VGPRs per lane gives the half-row-block sums; one shfl_xor(16)
// combines the two halves so every lane holds the full 32-lane sum.
// Must be called with EXEC all ones (no divergence at call site).
__device__ __forceinline__ float wave_sum_wmma(float v) {
    v2f a; a.x = v;    a.y = 0.0f;
    v2f b; b.x = 1.0f; b.y = 1.0f;
    v8f c = {};
    v8f d = __builtin_amdgcn_wmma_f32_16x16x4_f32(
        /*neg_a=*/false, a, /*neg_b=*/false, b,
        /*c_mod=*/(short)0, c, /*reuse_a=*/false, /*reuse_b=*/false);
    float s = ((d[0] + d[1]) + (d[2] + d[3])) + ((d[4] + d[5]) + (d[6] + d[7]));
    s += __shfl_xor(s, 16, 32);
    return s;
}

__device__ __forceinline__ void acc_px(float l, float t,
                                       float& ps, float& pc,
                                       float& ns, float& nc) {
    float d  = l - t;
    float d2 = d * d;
    if (t > 0.0f)  { ps += d2; pc += 1.0f; }
    if (t == 0.0f) { ns += d2; nc += 1.0f; }
}

// Pass 1: each block reduces a contiguous 16384-element chunk of one plane.
// Writes {pos_sum, pos_cnt, neg_sum, neg_cnt} per block into ws.
__global__ __launch_bounds__(THREADS)
void smse_pass1(const float* __restrict__ logits,
                const float* __restrict__ targets,
                float* __restrict__ partials) {
    const int plane = blockIdx.x / SUBBLOCKS;
    const int sub   = blockIdx.x % SUBBLOCKS;
    const int base4 = plane * PLANE_F4 + sub * CHUNK_F4;

    const float4* __restrict__ l4 = (const float4*)logits;
    const float4* __restrict__ t4 = (const float4*)targets;

    float ps = 0.0f, pc = 0.0f, ns = 0.0f, nc = 0.0f;

#pragma unroll 4
    for (int i = threadIdx.x; i < CHUNK_F4; i += THREADS) {
        float4 lv = l4[base4 + i];
        float4 tv = t4[base4 + i];
        acc_px(lv.x, tv.x, ps, pc, ns, nc);
        acc_px(lv.y, tv.y, ps, pc, ns, nc);
        acc_px(lv.z, tv.z, ps, pc, ns, nc);
        acc_px(lv.w, tv.w, ps, pc, ns, nc);
    }

    // Wave-level reduction via the matrix pipe (full EXEC here, pre-divergence).
    float wps = wave_sum_wmma(ps);
    float wpc = wave_sum_wmma(pc);
    float wns = wave_sum_wmma(ns);
    float wnc = wave_sum_wmma(nc);

    __shared__ float sred[(THREADS / 32) * 4];
    const int lane = threadIdx.x & 31;
    const int wid  = threadIdx.x >> 5;
    if (lane == 0) {
        sred[wid * 4 + 0] = wps;
        sred[wid * 4 + 1] = wpc;
        sred[wid * 4 + 2] = wns;
        sred[wid * 4 + 3] = wnc;
    }
    __syncthreads();

    if (threadIdx.x < 4) {
        float s = 0.0f;
#pragma unroll
        for (int w = 0; w < THREADS / 32; ++w) s += sred[w * 4 + threadIdx.x];
        partials[(size_t)blockIdx.x * 4 + threadIdx.x] = s;
    }
}

// Pass 2: fold SUBBLOCKS partials per plane, apply masked-MSE semantics,
// sum over the 64 planes, divide by B*C.
__global__ __launch_bounds__(64)
void smse_pass2(const float* __restrict__ partials, float* __restrict__ out) {
    __shared__ float terms[PLANES];
    const int t = threadIdx.x;

    float term = 0.0f;
    if (t < PLANES) {
        float ps = 0.0f, pc = 0.0f, ns = 0.0f, nc = 0.0f;
        for (int s = 0; s < SUBBLOCKS; ++s) {
            const float* p = partials + (size_t)(t * SUBBLOCKS + s) * 4;
            ps += p[0]; pc += p[1]; ns += p[2]; nc += p[3];
        }
        term = ((pc > 0.0f) ? ps / pc : 0.0f) + ((nc > 0.0f) ? ns / nc : 0.0f);
    }
    terms[t] = term;
    __syncthreads();

    if (t == 0) {
        float tot = 0.0f;
#pragma unroll
        for (int i = 0; i < PLANES; ++i) tot += terms[i];
        out[0] = tot / (float)PLANES;  // / num_classes / batch_size
    }
}

extern "C" void kernel_launch(void* const* d_in, const int* in_sizes, int n_in,
                              void* d_out, int out_size, void* d_ws, size_t ws_size,
                              hipStream_t stream) {
    (void)in_sizes; (void)n_in; (void)out_size; (void)ws_size;
    const float* logits  = (const float*)d_in[0];
    const float* targets = (const float*)d_in[1];
    float* partials = (float*)d_ws;            // PLANES*SUBBLOCKS*4 floats = 16 KB
    float* out      = (float*)d_out;           // single f32 scalar

    smse_pass1<<<PLANES * SUBBLOCKS, THREADS, 0, stream>>>(logits, targets, partials);
    smse_pass2<<<1, 64, 0, stream>>>(partials, out);
}